// CandidateRepresentationLayer_85383949844587
// MI455X (gfx1250) — compile-verified
//
#include <hip/hip_runtime.h>
#include <hip/hip_bf16.h>

// Problem constants (fixed by the reference).
#define B_ 64
#define S_ 512
#define A_ 8
#define D_ 128
#define C_ 1024

// Flat f32 output offsets, in reference tuple return order.
#define OUT_REPR  0
#define OUT_LABEL (B_ * C_ * D_)            /* 8388608 */
#define OUT_NUM   (OUT_LABEL + B_ * C_)     /* 8454144 */
#define OUT_LEN   (OUT_NUM + 1)             /* 8454145 */
#define OUT_MASK  (OUT_LEN + B_)            /* 8454209 */
#define OUT_LOC   (OUT_MASK + B_ * C_)      /* 8519745 */

// ---- CDNA5 async global<->LDS DMA path (ASYNCcnt), guarded ----------------
#if defined(__has_builtin)
#if __has_builtin(__builtin_amdgcn_global_load_async_to_lds_b128) && \
    __has_builtin(__builtin_amdgcn_global_store_async_from_lds_b128) && \
    __has_builtin(__builtin_amdgcn_s_wait_asynccnt)
#define HAVE_ASYNC_LDS 1
#endif
#endif

#ifdef HAVE_ASYNC_LDS
typedef int v4i __attribute__((ext_vector_type(4)));
typedef __attribute__((address_space(1))) v4i gas_v4i;   // global (prints as __device__)
typedef __attribute__((address_space(3))) v4i las_v4i;   // LDS
#endif

// ---------------------------------------------------------------------------
// Kernel 1: labels / mask / loc / per-batch len / scalar C.
// One block per batch; 256 threads each handle 4 candidates.
// ---------------------------------------------------------------------------
__global__ __launch_bounds__(256) void
cand_small_outputs_kernel(const int* __restrict__ cls,
                          const int* __restrict__ loc,
                          const int* __restrict__ cand,
                          float* __restrict__ out) {
    const int b = blockIdx.x;
    const int t = threadIdx.x;
    __shared__ float red[256];

    float sum = 0.0f;
#pragma unroll
    for (int i = 0; i < C_ / 256; ++i) {
        const int c = t + i * 256;
        const int g = b * C_ + c;                 // global candidate id
        const int bb = cand[g * 3 + 0];
        const int w  = cand[g * 3 + 1];
        const int a  = cand[g * 3 + 2];
        const int ai = (bb * S_ + w) * A_ + a;    // flat anchor index

        const int lbl = cls[ai];
        const int l0  = loc[ai * 2 + 0];
        const int l1  = loc[ai * 2 + 1];
        const float m = (l0 != l1) ? 1.0f : 0.0f;
        sum += m;

        out[OUT_LABEL + g]       = (float)lbl;
        out[OUT_MASK  + g]       = m;
        out[OUT_LOC + g * 2 + 0] = (float)l0;
        out[OUT_LOC + g * 2 + 1] = (float)l1;
    }

    red[t] = sum;
    __syncthreads();
#pragma unroll
    for (int s = 128; s > 0; s >>= 1) {
        if (t < s) red[t] += red[t + s];
        __syncthreads();
    }
    if (t == 0) {
        float len = red[0];
        if (len < 1.0f) len = 1.0f;
        out[OUT_LEN + b] = len;
        if (b == 0) out[OUT_NUM] = (float)C_;
    }
}

// ---------------------------------------------------------------------------
// Kernel 2: masked row gather of word_repr.
// 256 threads = 8 waves; each wave copies 8 rows of 512 B (lane = one float4).
// Unmasked rows: CDNA5 async DMA global->LDS->global (no VGPR data movement),
// 8 loads in flight before a single s_wait_asynccnt 0. Masked rows: store 0.
// ---------------------------------------------------------------------------
#define ROWS_PER_WAVE 8

__global__ __launch_bounds__(256) void
repr_gather_kernel(const float* __restrict__ wr,
                   const int* __restrict__ loc,
                   const int* __restrict__ cand,
                   float* __restrict__ out) {
    __shared__ float lds[8 * ROWS_PER_WAVE * D_];   // 8 waves * 8 rows * 512B = 32 KB

    const int t    = threadIdx.x;
    const int wave = t >> 5;
    const int lane = t & 31;
    const int r0   = (blockIdx.x * 8 + wave) * ROWS_PER_WAVE;

    // gfx1250 prefetch of this wave's index records (codegen: global_prefetch_b8)
    __builtin_prefetch(&cand[(size_t)r0 * 3], 0, 0);

    float* slot = &lds[(wave * ROWS_PER_WAVE) * D_];
    unsigned maskbits = 0;
#ifndef HAVE_ASYNC_LDS
    float4 regs[ROWS_PER_WAVE];
#endif

    // Phase 1: issue all row loads (async DMA or VGPR), record mask bits.
#pragma unroll
    for (int i = 0; i < ROWS_PER_WAVE; ++i) {
        const int g  = r0 + i;
        const int bb = cand[g * 3 + 0];
        const int w  = cand[g * 3 + 1];
        const int a  = cand[g * 3 + 2];
        const long ai = (long)(bb * S_ + w) * A_ + a;
        const int l0 = loc[ai * 2 + 0];
        const int l1 = loc[ai * 2 + 1];
        if (l0 != l1) {                       // wave-uniform: EXEC all-ones here
            maskbits |= (1u << i);
            const float* src = wr + ai * D_ + lane * 4;
#ifdef HAVE_ASYNC_LDS
            __builtin_amdgcn_global_load_async_to_lds_b128(
                (gas_v4i*)src, (las_v4i*)(slot + i * D_ + lane * 4), 0, 0);
#else
            regs[i] = *(const float4*)src;
#endif
        }
    }

    // Phase 2: one wait for all 8 in-flight row loads.
#ifdef HAVE_ASYNC_LDS
    __builtin_amdgcn_s_wait_asynccnt(0);
#endif

    // Phase 3: write rows (async store from LDS, or VGPR store); zeros if masked.
#pragma unroll
    for (int i = 0; i < ROWS_PER_WAVE; ++i) {
        float* dst = out + OUT_REPR + (long)(r0 + i) * D_ + lane * 4;
        if (maskbits & (1u << i)) {
#ifdef HAVE_ASYNC_LDS
            __builtin_amdgcn_global_store_async_from_lds_b128(
                (gas_v4i*)dst, (las_v4i*)(slot + i * D_ + lane * 4), 0, 0);
#else
            *(float4*)dst = regs[i];
#endif
        } else {
            *(float4*)dst = make_float4(0.0f, 0.0f, 0.0f, 0.0f);
        }
    }
    // S_ENDPGM performs an implicit S_WAIT_IDLE (all counters, incl. ASYNCcnt).
}

// ---------------------------------------------------------------------------
extern "C" void kernel_launch(void* const* d_in, const int* in_sizes, int n_in,
                              void* d_out, int out_size, void* d_ws, size_t ws_size,
                              hipStream_t stream) {
    (void)in_sizes; (void)n_in; (void)out_size; (void)d_ws; (void)ws_size;

    const float* wr  = (const float*)d_in[0];   // word_repr   (B,S,A,D) f32
    const int*   cls = (const int*)d_in[1];     // anchor_cls  (B,S,A)   i32
    const int*   loc = (const int*)d_in[2];     // anchor_loc  (B,S,A,2) i32
    const int*   cnd = (const int*)d_in[3];     // candidates_idx (B*C,3) i32
    float*       out = (float*)d_out;

    cand_small_outputs_kernel<<<B_, 256, 0, stream>>>(cls, loc, cnd, out);

    const int rows_per_block = 8 * ROWS_PER_WAVE;           // 64
    const int nblocks = (B_ * C_) / rows_per_block;         // 1024
    repr_gather_kernel<<<nblocks, 256, 0, stream>>>(wr, loc, cnd, out);
}